// Baseline_83837761618106
// MI455X (gfx1250) — compile-verified
//
#include <hip/hip_runtime.h>
#include <hip/hip_bf16.h>

typedef __attribute__((ext_vector_type(2))) float v2f;
typedef __attribute__((ext_vector_type(8))) float v8f;

#define BATCHES 64
#define VOX     64          // 4^3
#define NCLASS  40
#define NPAD    48          // 40 padded to 3 x 16 tiles
#define HIST_THREADS 256
#define HIST_WAVES   (HIST_THREADS / 32)
#define BLOCKS_PER_BATCH 32

// ---------------------------------------------------------------------------
// Zero the per-batch voxel counters in workspace.
// ---------------------------------------------------------------------------
__global__ void zero_counts_kernel(unsigned int* __restrict__ counts, int n) {
    int i = blockIdx.x * blockDim.x + threadIdx.x;
    if (i < n) counts[i] = 0u;
}

// ---------------------------------------------------------------------------
// Bin one point into the wave-private LDS histogram.
// ---------------------------------------------------------------------------
__device__ __forceinline__ void bin_point(float px, float py, float pz,
                                          unsigned int* __restrict__ whist) {
    bool inside = (px >= -2.0f) && (px <= 2.0f) &&
                  (py >= -2.0f) && (py <= 2.0f) &&
                  (pz >= -2.0f) && (pz <= 2.0f);
    if (inside) {
        int ix = (int)floorf(px + 2.0f); ix = ix < 0 ? 0 : (ix > 3 ? 3 : ix);
        int iy = (int)floorf(py + 2.0f); iy = iy < 0 ? 0 : (iy > 3 ? 3 : iy);
        int iz = (int)floorf(pz + 2.0f); iz = iz < 0 ? 0 : (iz > 3 ? 3 : iz);
        atomicAdd(&whist[(ix << 4) | (iy << 2) | iz], 1u);
    }
}

// ---------------------------------------------------------------------------
// Streaming voxel histogram. 4 points per thread-iteration via 3x b128 loads
// (16B-aligned by construction), per-wave replicated LDS histograms (wave32
// contention containment), then one global atomic per bin per block.
// ---------------------------------------------------------------------------
__global__ void voxel_hist_kernel(const float* __restrict__ x,
                                  unsigned int* __restrict__ counts,
                                  int nPoints, int perBlock) {
    __shared__ unsigned int lhist[HIST_WAVES][VOX];

    const int tid  = threadIdx.x;
    const int wave = tid >> 5;          // wave32 on gfx1250
    unsigned int* whist = &lhist[wave][0];

    for (int i = tid; i < HIST_WAVES * VOX; i += HIST_THREADS)
        (&lhist[0][0])[i] = 0u;
    __syncthreads();

    const int batch      = blockIdx.x / BLOCKS_PER_BATCH;
    const int blkInBatch = blockIdx.x % BLOCKS_PER_BATCH;
    const int start      = blkInBatch * perBlock;          // multiple of 4
    const int end        = (start + perBlock < nPoints) ? (start + perBlock) : nPoints;

    const float* __restrict__ xb = x + (size_t)batch * (size_t)nPoints * 3u;

    // 4 points (48 bytes) per thread per iteration.
    for (int i = start + tid * 4; i < end; i += HIST_THREADS * 4) {
        const float* p = xb + (size_t)i * 3u;
        if (i + 4 <= end) {
            // i is a multiple of 4 -> byte offset multiple of 48 -> 16B aligned
            const float4* q = (const float4*)p;
            __builtin_prefetch(p + 12 * HIST_THREADS, 0, 1);   // next group
            float4 q0 = q[0];
            float4 q1 = q[1];
            float4 q2 = q[2];
            bin_point(q0.x, q0.y, q0.z, whist);
            bin_point(q0.w, q1.x, q1.y, whist);
            bin_point(q1.z, q1.w, q2.x, whist);
            bin_point(q2.y, q2.z, q2.w, whist);
        } else {
            for (int j = 0; i + j < end; ++j)
                bin_point(p[3 * j], p[3 * j + 1], p[3 * j + 2], whist);
        }
    }
    __syncthreads();

    for (int bin = tid; bin < VOX; bin += HIST_THREADS) {
        unsigned int s = 0;
        #pragma unroll
        for (int w = 0; w < HIST_WAVES; ++w) s += lhist[w][bin];
        if (s) atomicAdd(&counts[batch * VOX + bin], s);
    }
}

// ---------------------------------------------------------------------------
// Normalize counts and compute out[64,40] = counts @ W^T + b with
// V_WMMA_F32_16X16X4_F32 (full fp32 path). 1 block, 4 waves; each wave owns
// one 16-row M tile and iterates 3 N tiles x 16 K-steps (K=4 each).
// ---------------------------------------------------------------------------
__global__ void finalize_gemm_kernel(const unsigned int* __restrict__ counts,
                                     const float* __restrict__ W,
                                     const float* __restrict__ bias,
                                     float* __restrict__ out) {
    __shared__ float cnt[BATCHES][VOX];   // 16 KB, normalized counts (A)
    __shared__ float Ws[NPAD][VOX];       // 12 KB, W zero-padded to 48 rows
    __shared__ float bs[NPAD];
    __shared__ float inv[BATCHES];

    const int tid = threadIdx.x;          // 128 threads = 4 waves

    for (int i = tid; i < BATCHES * VOX; i += blockDim.x)
        (&cnt[0][0])[i] = (float)counts[i];
    for (int i = tid; i < NPAD * VOX; i += blockDim.x) {
        int r = i >> 6, c = i & 63;
        (&Ws[0][0])[i] = (r < NCLASS) ? W[r * VOX + c] : 0.0f;
    }
    for (int i = tid; i < NPAD; i += blockDim.x)
        bs[i] = (i < NCLASS) ? bias[i] : 0.0f;
    __syncthreads();

    if (tid < BATCHES) {
        float s = 0.0f;
        #pragma unroll
        for (int j = 0; j < VOX; ++j) s += cnt[tid][j];
        inv[tid] = (s > 0.0f) ? (1.0f / s) : 0.0f;
    }
    __syncthreads();

    for (int i = tid; i < BATCHES * VOX; i += blockDim.x)
        (&cnt[0][0])[i] *= inv[i >> 6];
    __syncthreads();

    const int wave   = tid >> 5;
    const int lane   = tid & 31;
    const int laneLo = lane & 15;
    const int hiHalf = lane >> 4;         // 0: K pair {0,1}, 1: K pair {2,3}
    const int mbase  = wave * 16;
    const int arow   = mbase + laneLo;

    for (int nt = 0; nt < 3; ++nt) {
        const int nbase = nt * 16;
        const int bcol  = nbase + laneLo;
        v8f acc = {};
        #pragma unroll
        for (int k = 0; k < 16; ++k) {
            const int k0 = k * 4 + hiHalf * 2;
            v2f a, b;
            a.x = cnt[arow][k0];
            a.y = cnt[arow][k0 + 1];
            b.x = Ws[bcol][k0];           // B[k][n] = W^T[k][n] = W[n][k]
            b.y = Ws[bcol][k0 + 1];
            acc = __builtin_amdgcn_wmma_f32_16x16x4_f32(
                /*neg_a=*/false, a, /*neg_b=*/false, b,
                /*c_mod=*/(short)0, acc, /*reuse_a=*/false, /*reuse_b=*/false);
        }
        // D layout: VGPR j, lane L -> row = mbase + j + (L>=16 ? 8 : 0)
        const int col = nbase + laneLo;
        if (col < NCLASS) {
            const float bb = bs[col];
            #pragma unroll
            for (int j = 0; j < 8; ++j) {
                const int row = mbase + j + hiHalf * 8;
                out[row * NCLASS + col] = acc[j] + bb;
            }
        }
    }
}

// ---------------------------------------------------------------------------
extern "C" void kernel_launch(void* const* d_in, const int* in_sizes, int n_in,
                              void* d_out, int out_size, void* d_ws, size_t ws_size,
                              hipStream_t stream) {
    const float* x  = (const float*)d_in[0];   // [64, N, 3]
    const float* W  = (const float*)d_in[1];   // [40, 64]
    const float* bv = (const float*)d_in[2];   // [40]
    float* out      = (float*)d_out;           // [64, 40]
    unsigned int* counts = (unsigned int*)d_ws; // [64 * 64]

    const int nPoints = in_sizes[0] / (BATCHES * 3);
    // per-block chunk: round up to a multiple of 4 so every thread group of
    // 4 points starts 16B-aligned (point index multiple of 4 -> offset % 48 == 0)
    int perBlock = (nPoints + BLOCKS_PER_BATCH - 1) / BLOCKS_PER_BATCH;
    perBlock = (perBlock + 3) & ~3;

    zero_counts_kernel<<<(BATCHES * VOX + 255) / 256, 256, 0, stream>>>(
        counts, BATCHES * VOX);

    voxel_hist_kernel<<<BATCHES * BLOCKS_PER_BATCH, HIST_THREADS, 0, stream>>>(
        x, counts, nPoints, perBlock);

    finalize_gemm_kernel<<<1, 128, 0, stream>>>(counts, W, bv, out);
}